// SGC_927712936483
// MI455X (gfx1250) — compile-verified
//
#include <hip/hip_runtime.h>
#include <hip/hip_bf16.h>

typedef __attribute__((ext_vector_type(2))) float v2f;
typedef __attribute__((ext_vector_type(8))) float v8f;

// ---------------------------------------------------------------------------
// Zero a float buffer (workspace is poisoned 0xAA by harness; we must clear
// the accumulation target before every scatter layer).
// ---------------------------------------------------------------------------
__global__ void sgc_zero_f32(float* __restrict__ p, int n) {
    int i = blockIdx.x * blockDim.x + threadIdx.x;
    if (i < n) p[i] = 0.0f;
}

// ---------------------------------------------------------------------------
// One SpMM layer: out[dst] += w * in[src], feature dim = 64.
// Thread layout: 16 threads per edge, each handles 4 consecutive features
// (float4 load from L2-resident H, 4 float atomics into out).
// ---------------------------------------------------------------------------
__global__ void sgc_spmm_scatter(const int* __restrict__ src,
                                 const int* __restrict__ dst,
                                 const float* __restrict__ ew,
                                 const float* __restrict__ hin,
                                 float* __restrict__ hout,
                                 int E) {
    int tid = blockIdx.x * blockDim.x + threadIdx.x;
    int e = tid >> 4;
    if (e >= E) return;
    int q = (tid & 15) * 4;

    int s = src[e];
    int t = dst[e];
    float w = ew[e];

    const float4 hv = *reinterpret_cast<const float4*>(hin + (size_t)s * 64 + q);
    float* o = hout + (size_t)t * 64 + q;
    atomicAdd(o + 0, w * hv.x);
    atomicAdd(o + 1, w * hv.y);
    atomicAdd(o + 2, w * hv.z);
    atomicAdd(o + 3, w * hv.w);
}

// ---------------------------------------------------------------------------
// Dense linear: out[N,64] = H[N,64] @ W[64,64] + b, fp32 WMMA.
// One wave per 16x16 output tile; K-loop over 64 in steps of 4 using
// V_WMMA_F32_16X16X4_F32 (full fp32 precision, matches reference).
//
// VGPR layouts (ISA 7.12.2, wave32):
//   A 16x4:  lanes 0-15 -> M=lane, K = {k0,k0+1}; lanes 16-31 -> K = {k0+2,k0+3}
//   B 4x16:  vgpr0 -> row K = k0+2*half, vgpr1 -> K = k0+2*half+1, N = lane%16
//   C/D 16x16: vgpr v -> M = v + 8*half, N = lane%16
// ---------------------------------------------------------------------------
__global__ void sgc_gemm_wmma(const float* __restrict__ H,
                              const float* __restrict__ W,
                              const float* __restrict__ b,
                              float* __restrict__ out,
                              int nrows) {
    const int wave = (blockIdx.x * blockDim.x + threadIdx.x) >> 5;
    const int lane = threadIdx.x & 31;
    const int mt = wave >> 2;        // row-tile index (64 cols = 4 col tiles)
    const int nt = wave & 3;         // col-tile index
    const int m0 = mt * 16;
    if (m0 >= nrows) return;         // wave-uniform: EXEC stays all-1s for WMMA
    const int n0 = nt * 16;

    const int half = lane >> 4;      // 0: lanes 0-15, 1: lanes 16-31
    const int l16 = lane & 15;

    // C init: every output element gets + b[col]
    v8f c;
    const float bias = b[n0 + l16];
#pragma unroll
    for (int v = 0; v < 8; ++v) c[v] = bias;

#pragma unroll
    for (int k0 = 0; k0 < 64; k0 += 4) {
        // A-matrix fragment: row m0+l16, K pair selected by half
        const float* arow = H + (size_t)(m0 + l16) * 64 + k0 + half * 2;
        v2f a;
        a.x = arow[0];
        a.y = arow[1];

        // B-matrix fragment: rows k0+2*half (+1), column n0+l16
        const float* wrow = W + (size_t)(k0 + half * 2) * 64 + n0 + l16;
        v2f bb;
        bb.x = wrow[0];
        bb.y = wrow[64];

        // 8 args: (neg_a, A, neg_b, B, c_mod, C, reuse_a, reuse_b)
        c = __builtin_amdgcn_wmma_f32_16x16x4_f32(
            false, a, false, bb, (short)0, c, false, false);
    }

#pragma unroll
    for (int v = 0; v < 8; ++v) {
        out[(size_t)(m0 + v + half * 8) * 64 + n0 + l16] = c[v];
    }
}

// ---------------------------------------------------------------------------
// Driver: 3x (zero + scatter), then WMMA linear.
// d_in order: edge_index [2,E] i32, edge_weight [E] f32, H [N,64] f32,
//             W [64,64] f32, b [64] f32
// ---------------------------------------------------------------------------
extern "C" void kernel_launch(void* const* d_in, const int* in_sizes, int n_in,
                              void* d_out, int out_size, void* d_ws, size_t ws_size,
                              hipStream_t stream) {
    const int*   edge_index = (const int*)d_in[0];
    const float* ew         = (const float*)d_in[1];
    const float* H          = (const float*)d_in[2];
    const float* W          = (const float*)d_in[3];
    const float* b          = (const float*)d_in[4];
    float*       out        = (float*)d_out;

    const int E = in_sizes[1];          // 800000
    const int N = in_sizes[2] / 64;     // 50000
    const int NF = N * 64;

    const int* src = edge_index;        // edge_index[0, :]
    const int* dst = edge_index + E;    // edge_index[1, :]

    float* tmp0 = (float*)d_ws;
    float* tmp1 = tmp0 + (size_t)NF;

    const int BLK = 256;
    const int zeroGrid = (NF + BLK - 1) / BLK;
    const int spmmGrid = (E * 16 + BLK - 1) / BLK;

    // Layer 1: tmp0 = A @ H
    sgc_zero_f32<<<zeroGrid, BLK, 0, stream>>>(tmp0, NF);
    sgc_spmm_scatter<<<spmmGrid, BLK, 0, stream>>>(src, dst, ew, H, tmp0, E);
    // Layer 2: tmp1 = A @ tmp0
    sgc_zero_f32<<<zeroGrid, BLK, 0, stream>>>(tmp1, NF);
    sgc_spmm_scatter<<<spmmGrid, BLK, 0, stream>>>(src, dst, ew, tmp0, tmp1, E);
    // Layer 3: tmp0 = A @ tmp1  (stream order guarantees tmp0 reads finished)
    sgc_zero_f32<<<zeroGrid, BLK, 0, stream>>>(tmp0, NF);
    sgc_spmm_scatter<<<spmmGrid, BLK, 0, stream>>>(src, dst, ew, tmp1, tmp0, E);

    // Linear: out = tmp0 @ W + b   (one wave per 16x16 tile)
    const int tilesM = (N + 15) / 16;
    const int wavesTotal = tilesM * 4;
    const int wavesPerBlk = BLK / 32;
    const int gemmGrid = (wavesTotal + wavesPerBlk - 1) / wavesPerBlk;
    sgc_gemm_wmma<<<gemmGrid, BLK, 0, stream>>>(tmp0, W, b, out, N);
}